// StochasticDQNModel_51951924412906
// MI455X (gfx1250) — compile-verified
//
#include <hip/hip_runtime.h>

typedef __attribute__((ext_vector_type(2))) float v2f;
typedef __attribute__((ext_vector_type(8))) float v8f;

// Async global->LDS (gfx1250, tracked by ASYNCcnt). Guarded so the file still
// compiles if this toolchain doesn't declare the builtin.
#if defined(__has_builtin)
#if __has_builtin(__builtin_amdgcn_global_load_async_to_lds_b128)
#define USE_ASYNC_LDS 1
#endif
#endif
#ifndef USE_ASYNC_LDS
#define USE_ASYNC_LDS 0
#endif

// ---------------------------------------------------------------- utilities

__global__ void k_zero(float* __restrict__ p, long n) {
  long i = (long)blockIdx.x * blockDim.x + threadIdx.x;
  if (i < n) p[i] = 0.0f;
}

// deg[c] += 1 per edge (float so we can reuse buffer for dinv)
__global__ void k_degree(const long long* __restrict__ col,
                         float* __restrict__ deg, int E) {
  int e = blockIdx.x * blockDim.x + threadIdx.x;
  if (e < E) atomicAdd(&deg[(int)col[e]], 1.0f);
}

__global__ void k_dinv(float* __restrict__ d, int n) {
  int i = blockIdx.x * blockDim.x + threadIdx.x;
  if (i < n) d[i] = rsqrtf(d[i] + 1.0f);
}

// s[c] += dinv[r] * x[r]  (layer-1 rank-1 scalar aggregation)
__global__ void k_edge_scalar(const long long* __restrict__ row,
                              const long long* __restrict__ col,
                              const float* __restrict__ x,
                              const float* __restrict__ dinv,
                              float* __restrict__ s, int E) {
  int e = blockIdx.x * blockDim.x + threadIdx.x;
  if (e < E) {
    int r = (int)row[e], c = (int)col[e];
    atomicAdd(&s[c], dinv[r] * x[r]);
  }
}

// h1[i][j] = relu((dinv[i]*s[i] + dinv[i]^2*x[i]) * W1[j] + b1[j])
__global__ void k_h1(const float* __restrict__ x, const float* __restrict__ s,
                     const float* __restrict__ dinv,
                     const float* __restrict__ W1, const float* __restrict__ b1,
                     float* __restrict__ h1, int n) {
  long t = (long)blockIdx.x * blockDim.x + threadIdx.x;
  int i = (int)(t >> 5);
  if (i >= n) return;
  int j = (int)(t & 31) * 4;
  float di = dinv[i];
  float base = di * s[i] + di * di * x[i];
  float4 w = *(const float4*)(W1 + j);
  float4 b = *(const float4*)(b1 + j);
  float4 h;
  h.x = fmaxf(base * w.x + b.x, 0.0f);
  h.y = fmaxf(base * w.y + b.y, 0.0f);
  h.z = fmaxf(base * w.z + b.z, 0.0f);
  h.w = fmaxf(base * w.w + b.w, 0.0f);
  *(float4*)(h1 + (size_t)i * 128 + j) = h;
}

// ---------------------------------------------------------------- WMMA GEMM
// g = h1 @ W2   [n,128] x [128,128], full f32 via V_WMMA_F32_16X16X4_F32.
// One wave computes a 16x128 strip: 8 accumulator tiles, K looped in steps
// of 4. W2 staged in LDS (64 KB) once per workgroup — via async global->LDS
// (GLOBAL_LOAD_ASYNC_TO_LDS_B128 + s_wait_asynccnt) when available.
__global__ __launch_bounds__(256) void k_gemm(const float* __restrict__ A,
                                              const float* __restrict__ B,
                                              float* __restrict__ C, int n) {
  __shared__ float sB[128 * 128];
#if USE_ASYNC_LDS
  {
    typedef int b128 __attribute__((vector_size(16)));
    typedef __attribute__((address_space(1))) b128* gp;
    typedef __attribute__((address_space(3))) b128* lp;
    gp gsrc = (gp)B;
    lp ldst = (lp)sB;
    for (int idx = threadIdx.x; idx < 128 * 128 / 4; idx += blockDim.x)
      __builtin_amdgcn_global_load_async_to_lds_b128(gsrc + idx, ldst + idx, 0, 0);
    asm volatile("s_wait_asynccnt 0x0" ::: "memory");
  }
#else
  {
    const float4* src = (const float4*)B;
    float4* dst = (float4*)sB;
    for (int idx = threadIdx.x; idx < 128 * 128 / 4; idx += blockDim.x)
      dst[idx] = src[idx];
  }
#endif
  __syncthreads();

  int lane = threadIdx.x & 31;
  int strip = blockIdx.x * 8 + (threadIdx.x >> 5);
  int row0 = strip * 16;
  if (row0 >= n) return;

  if (row0 + 16 <= n) {
    // full-strip WMMA path (wave-uniform branch -> EXEC all ones)
    int mc = lane & 15;            // row (A) / col (B,C) within tile
    int kg = (lane >> 4) << 1;     // 0 or 2: which K pair this half-wave holds
    int vrow = (lane >> 4) << 3;   // +8 row offset for upper half-wave in C

    v8f acc[8] = {};               // zero accumulators

    const float* Arow = A + ((size_t)row0 + mc) * 128;
#pragma unroll 2
    for (int k = 0; k < 128; k += 4) {
      v2f a = *(const v2f*)(Arow + k + kg);   // A[m][k+kg], A[m][k+kg+1]
#pragma unroll
      for (int nt = 0; nt < 8; ++nt) {
        int ncol = nt * 16 + mc;
        v2f b;
        b.x = sB[(k + kg) * 128 + ncol];
        b.y = sB[(k + kg + 1) * 128 + ncol];
        acc[nt] = __builtin_amdgcn_wmma_f32_16x16x4_f32(
            false, a, false, b, (short)0, acc[nt], false, false);
      }
    }
#pragma unroll
    for (int nt = 0; nt < 8; ++nt) {
#pragma unroll
      for (int v = 0; v < 8; ++v) {
        C[(size_t)(row0 + vrow + v) * 128 + nt * 16 + mc] = acc[nt][v];
      }
    }
  } else {
    // scalar tail (never taken for n % 16 == 0, kept for safety)
    for (int r = row0; r < n; ++r) {
      for (int c = lane; c < 128; c += 32) {
        float acc = 0.0f;
        for (int k = 0; k < 128; ++k)
          acc += A[(size_t)r * 128 + k] * sB[k * 128 + c];
        C[(size_t)r * 128 + c] = acc;
      }
    }
  }
}

// ---------------------------------------------------------------- layer-2 edges
// agg[c][j] += dinv[r] * g[r][j]   (one wave per edge, float4 per lane)
__global__ void k_edge_vec(const long long* __restrict__ row,
                           const long long* __restrict__ col,
                           const float* __restrict__ dinv,
                           const float* __restrict__ g,
                           float* __restrict__ agg, int E) {
  long t = (long)blockIdx.x * blockDim.x + threadIdx.x;
  int e = (int)(t >> 5);
  if (e >= E) return;
  int l = (int)(t & 31) * 4;
  int r = (int)row[e], c = (int)col[e];
  float w = dinv[r];
  float4 gv = *(const float4*)(g + (size_t)r * 128 + l);
  float* dst = agg + (size_t)c * 128 + l;
  atomicAdd(dst + 0, w * gv.x);
  atomicAdd(dst + 1, w * gv.y);
  atomicAdd(dst + 2, w * gv.z);
  atomicAdd(dst + 3, w * gv.w);
}

// ---------------------------------------------------------------- epilogue
// h2 = relu(dinv[i]*agg + dinv[i]^2*g + b2); out = h2 @ Wl^T + bl
// one wave per node; 4 channels per lane; shuffle-reduce across 32 lanes.
__global__ void k_out(const float* __restrict__ agg, const float* __restrict__ g,
                      const float* __restrict__ dinv,
                      const float* __restrict__ b2, const float* __restrict__ Wl,
                      const float* __restrict__ bl, float* __restrict__ out,
                      int n) {
  long t = (long)blockIdx.x * blockDim.x + threadIdx.x;
  int i = (int)(t >> 5);
  if (i >= n) return;
  int lane = (int)(t & 31);
  int l = lane * 4;
  float di = dinv[i];
  float d2 = di * di;
  float4 a = *(const float4*)(agg + (size_t)i * 128 + l);
  float4 gv = *(const float4*)(g + (size_t)i * 128 + l);
  float4 b = *(const float4*)(b2 + l);
  float4 h;
  h.x = fmaxf(di * a.x + d2 * gv.x + b.x, 0.0f);
  h.y = fmaxf(di * a.y + d2 * gv.y + b.y, 0.0f);
  h.z = fmaxf(di * a.z + d2 * gv.z + b.z, 0.0f);
  h.w = fmaxf(di * a.w + d2 * gv.w + b.w, 0.0f);
#pragma unroll
  for (int o = 0; o < 4; ++o) {
    float4 w = *(const float4*)(Wl + o * 128 + l);
    float p = h.x * w.x + h.y * w.y + h.z * w.z + h.w * w.w;
    p += __shfl_xor(p, 16);
    p += __shfl_xor(p, 8);
    p += __shfl_xor(p, 4);
    p += __shfl_xor(p, 2);
    p += __shfl_xor(p, 1);
    if (lane == 0) out[(size_t)i * 4 + o] = p + bl[o];
  }
}

// ---------------------------------------------------------------- launch

extern "C" void kernel_launch(void* const* d_in, const int* in_sizes, int n_in,
                              void* d_out, int out_size, void* d_ws, size_t ws_size,
                              hipStream_t stream) {
  const float* x       = (const float*)d_in[0];
  const long long* edg = (const long long*)d_in[1];   // int64 [2, E]
  const float* W1      = (const float*)d_in[2];
  const float* b1      = (const float*)d_in[3];
  const float* W2      = (const float*)d_in[4];
  const float* b2      = (const float*)d_in[5];
  const float* Wl      = (const float*)d_in[6];
  const float* bl      = (const float*)d_in[7];
  float* out           = (float*)d_out;

  int n = in_sizes[0];          // N (x is [N,1])
  int E = in_sizes[1] / 2;
  const long long* row = edg;
  const long long* col = edg + E;

  // workspace: dinv[n] | s[n] | bufA[n*128] (h1 -> agg2) | bufB[n*128] (g)
  float* dinv = (float*)d_ws;
  float* s    = dinv + n;
  float* bufA = s + n;
  float* bufB = bufA + (size_t)n * 128;

  const int T = 256;
  // degrees -> dinv
  k_zero<<<(n + T - 1) / T, T, 0, stream>>>(dinv, n);
  k_degree<<<(E + T - 1) / T, T, 0, stream>>>(col, dinv, E);
  k_dinv<<<(n + T - 1) / T, T, 0, stream>>>(dinv, n);

  // layer 1 (rank-1): scalar edge aggregation, then expand to 128 + relu
  k_zero<<<(n + T - 1) / T, T, 0, stream>>>(s, n);
  k_edge_scalar<<<(E + T - 1) / T, T, 0, stream>>>(row, col, x, dinv, s, E);
  long th1 = (long)n * 32;
  k_h1<<<(unsigned)((th1 + T - 1) / T), T, 0, stream>>>(x, s, dinv, W1, b1, bufA, n);

  // g = h1 @ W2  (WMMA f32)
  int strips = (n + 15) / 16;
  k_gemm<<<(strips + 7) / 8, T, 0, stream>>>(bufA, W2, bufB, n);

  // layer 2 edge aggregation into bufA (reused)
  long big = (long)n * 128;
  k_zero<<<(unsigned)((big + T - 1) / T), T, 0, stream>>>(bufA, big);
  long tev = (long)E * 32;
  k_edge_vec<<<(unsigned)((tev + T - 1) / T), T, 0, stream>>>(row, col, dinv, bufB, bufA, E);

  // fused: bias + relu + linear head
  k_out<<<(unsigned)((th1 + T - 1) / T), T, 0, stream>>>(bufA, bufB, dinv, b2, Wl, bl, out, n);
}